// DictionaryLearning_90890097918488
// MI455X (gfx1250) — compile-verified
//
#include <hip/hip_runtime.h>

typedef float v2f __attribute__((ext_vector_type(2)));
typedef float v8f __attribute__((ext_vector_type(8)));

#define N_ATOMS   1024
#define N_FEAT    512
#define K_MAX     12
#define BATCH_N   2048
#define DIAG_EPS  1e-4f
#define OMP_THREADS 256

// ---------------------------------------------------------------------------
// C[M,N] = A[M,K] * B[N,K]^T  (+ diag_eps on the global diagonal)
// One wave32 per 16x16 output tile, V_WMMA_F32_16X16X4_F32 over K in steps of 4.
// A 16x4 f32 layout: lane l -> (M = l%16, K = 2*(l>=16)) in v[0], K+1 in v[1].
// B 4x16 layout mirrors A with N = l%16. C/D: lanes 0-15 M=v, 16-31 M=v+8.
// ---------------------------------------------------------------------------
__global__ __launch_bounds__(32)
void gemm_nt_wmma_f32(const float* __restrict__ A, const float* __restrict__ B,
                      float* __restrict__ C, int N, int K, float diag_eps)
{
    const int lane = threadIdx.x;          // 0..31, full wave => EXEC all ones
    const int r    = lane & 15;
    const int koff = (lane >> 4) << 1;     // 0 or 2
    const int tile_n = blockIdx.x;
    const int tile_m = blockIdx.y;

    const float* a = A + (size_t)(tile_m * 16 + r) * K + koff;
    const float* b = B + (size_t)(tile_n * 16 + r) * K + koff;

    v8f c = {};
    for (int k = 0; k < K; k += 4) {
        v2f av = *(const v2f*)(a + k);
        v2f bv = *(const v2f*)(b + k);
        // 8 args: (neg_a, A, neg_b, B, c_mod, C, reuse_a, reuse_b)
        c = __builtin_amdgcn_wmma_f32_16x16x4_f32(false, av, false, bv,
                                                  (short)0, c, false, false);
    }

    const int n  = tile_n * 16 + r;
    const int m0 = tile_m * 16 + ((lane >> 4) << 3);
#pragma unroll
    for (int v = 0; v < 8; ++v) {
        const int m = m0 + v;
        float val = c[v];
        if (m == n) val += diag_eps;
        C[(size_t)m * N + n] = val;
    }
}

// ---------------------------------------------------------------------------
// Batched OMP iterations: one workgroup (256 threads = 8 wave32) per signal.
// h / h_bar rows and all *selected* G rows live in LDS (CDNA5: 320KB/WGP).
// Each selected atom's G row is copied to LDS exactly once; every later
// access (column gathers on the serial lane, the beta rank-k update) is an
// LDS read with consecutive-bank addressing. Global G traffic per signal
// drops from ~78 rows to exactly 12.
// ---------------------------------------------------------------------------
__global__ __launch_bounds__(OMP_THREADS)
void omp_iterate(const float* __restrict__ hbar_g,   // [B, N_ATOMS]
                 const float* __restrict__ G,        // [N_ATOMS, N_ATOMS]
                 const float* __restrict__ Dm,       // [N_ATOMS, N_FEAT]
                 float* __restrict__ out_ypred,      // [B, N_FEAT]
                 float* __restrict__ out_x)          // [B, N_ATOMS]
{
    __shared__ float sG[K_MAX][N_ATOMS]; // cached selected G rows (48 KB)
    __shared__ float sh[N_ATOMS];        // current residual correlations h
    __shared__ float shbar[N_ATOMS];     // h_bar row
    __shared__ unsigned char smask[N_ATOMS];
    __shared__ float sL[K_MAX * K_MAX];  // lower-triangular factor
    __shared__ float sx[K_MAX];          // x_sel
    __shared__ int   sI[K_MAX];          // selected atoms
    __shared__ float rv[OMP_THREADS];
    __shared__ int   ri[OMP_THREADS];

    const int b   = blockIdx.x;
    const int tid = threadIdx.x;
    const float* hb = hbar_g + (size_t)b * N_ATOMS;

    for (int n = tid; n < N_ATOMS; n += OMP_THREADS) {
        float v = hb[n];
        shbar[n] = v;
        sh[n]    = v;
        smask[n] = 0;
    }
    __syncthreads();

    for (int t = 0; t < K_MAX; ++t) {
        // ---- argmax over masked |h| (lowest index wins ties) ----
        float best = -1.0f; int bidx = 0;
        for (int n = tid; n < N_ATOMS; n += OMP_THREADS) {
            float v = smask[n] ? -1.0f : fabsf(sh[n]);
            if (v > best) { best = v; bidx = n; }
        }
        rv[tid] = best; ri[tid] = bidx;
        __syncthreads();
        for (int s = OMP_THREADS / 2; s > 0; s >>= 1) {
            if (tid < s) {
                float ov = rv[tid + s]; int oi = ri[tid + s];
                if (ov > rv[tid] || (ov == rv[tid] && oi < ri[tid])) {
                    rv[tid] = ov; ri[tid] = oi;
                }
            }
            __syncthreads();
        }
        const int idx = ri[0];

        // ---- stage G[idx, :] into LDS once (coalesced, reused forever) ----
        {
            const float* grow = G + (size_t)idx * N_ATOMS;
            for (int n = tid; n < N_ATOMS; n += OMP_THREADS)
                sG[t][n] = grow[n];
        }
        __syncthreads();

        // ---- tiny Cholesky update + triangular solves on lane 0 ----
        if (tid == 0) {
            smask[idx] = 1;
            if (t == 0) {
                sI[0] = idx;
                sL[0] = 1.0f;
            } else {
                float g[K_MAX], w[K_MAX];
                for (int k = 0; k < t; ++k) g[k] = sG[k][idx];   // G[sI[k], idx]
                for (int i = 0; i < t; ++i) {              // forward: L w = g
                    float s = g[i];
                    for (int j = 0; j < i; ++j) s -= sL[i * K_MAX + j] * w[j];
                    w[i] = s / sL[i * K_MAX + i];
                }
                float corner = sG[t][idx];                 // G[idx, idx] (has eps)
                for (int k = 0; k < t; ++k) corner -= w[k] * w[k];
                corner = sqrtf(fmaxf(corner, DIAG_EPS));
                for (int j = 0; j < t; ++j) sL[t * K_MAX + j] = w[j];
                sL[t * K_MAX + t] = corner;
                sI[t] = idx;
            }
            const int kk = t + 1;
            float y[K_MAX], xl[K_MAX];
            for (int i = 0; i < kk; ++i) {                 // forward: L y = h_sel
                float s = shbar[sI[i]];
                for (int j = 0; j < i; ++j) s -= sL[i * K_MAX + j] * y[j];
                y[i] = s / sL[i * K_MAX + i];
            }
            for (int i = kk - 1; i >= 0; --i) {            // backward: L^T x = y
                float s = y[i];
                for (int j = i + 1; j < kk; ++j) s -= sL[j * K_MAX + i] * xl[j];
                xl[i] = s / sL[i * K_MAX + i];
            }
            for (int i = 0; i < kk; ++i) sx[i] = xl[i];
        }
        __syncthreads();

        // ---- h = h_bar - x_sel . G[I]  (all reads from LDS) ----
        const int kk = t + 1;
        for (int n = tid; n < N_ATOMS; n += OMP_THREADS) {
            float acc = 0.0f;
            for (int k = 0; k < kk; ++k)
                acc = fmaf(sx[k], sG[k][n], acc);
            sh[n] = shbar[n] - acc;
        }
        __syncthreads();
    }

    // ---- X row: scatter x_sel at selected atoms, zeros elsewhere ----
    for (int n = tid; n < N_ATOMS; n += OMP_THREADS) {
        float v = 0.0f;
        for (int k = 0; k < K_MAX; ++k)
            if (sI[k] == n) v = sx[k];
        out_x[(size_t)b * N_ATOMS + n] = v;
    }

    // ---- Y_pred row = sum_k x_sel[k] * D[I_k, :] ----
    for (int m = tid; m < N_FEAT; m += OMP_THREADS) {
        float acc = 0.0f;
        for (int k = 0; k < K_MAX; ++k)
            acc = fmaf(sx[k], Dm[(size_t)sI[k] * N_FEAT + m], acc);
        out_ypred[(size_t)b * N_FEAT + m] = acc;
    }
}

extern "C" void kernel_launch(void* const* d_in, const int* in_sizes, int n_in,
                              void* d_out, int out_size, void* d_ws, size_t ws_size,
                              hipStream_t stream) {
    (void)in_sizes; (void)n_in; (void)out_size; (void)ws_size;
    const float* Y = (const float*)d_in[0];    // [2048, 512]
    const float* D = (const float*)d_in[1];    // [1024, 512]

    float* G    = (float*)d_ws;                       // [1024, 1024]  4 MB
    float* hbar = G + (size_t)N_ATOMS * N_ATOMS;      // [2048, 1024]  8 MB

    float* ypred = (float*)d_out;                     // [2048, 512]
    float* X     = ypred + (size_t)BATCH_N * N_FEAT;  // [2048, 1024]

    // G = D D^T + eps I
    gemm_nt_wmma_f32<<<dim3(N_ATOMS / 16, N_ATOMS / 16), dim3(32), 0, stream>>>(
        D, D, G, N_ATOMS, N_FEAT, DIAG_EPS);

    // h_bar = Y D^T
    gemm_nt_wmma_f32<<<dim3(N_ATOMS / 16, BATCH_N / 16), dim3(32), 0, stream>>>(
        Y, D, hbar, N_ATOMS, N_FEAT, 0.0f);

    // 12 greedy OMP steps per signal + outputs
    omp_iterate<<<dim3(BATCH_N), dim3(OMP_THREADS), 0, stream>>>(
        hbar, G, D, ypred, X);
}